// BlockRF_13357348290638
// MI455X (gfx1250) — compile-verified
//
#include <hip/hip_runtime.h>

typedef __attribute__((ext_vector_type(2))) float v2f;
typedef __attribute__((ext_vector_type(4))) float v4f;
typedef __attribute__((ext_vector_type(8))) float v8f;
typedef __attribute__((ext_vector_type(4))) int   v4i;

#define KH 3
#define KW 3
#define BSZ 32
#define HH 64
#define WW 64
#define CC 32
#define FF 32
#define OHN 62
#define OWN 62
#define KDIM 288          // KH*KW*CC
#define KCHUNK 144        // 2 chunks over K
#define AST 148           // ldsA row stride (floats): 20m mod 64 distinct for m=0..15 -> conflict-free b64 frag reads
#define WST 40            // ldsW row stride (floats): lane-half bank offset = +16 -> conflict-free b32 frag reads
#define NVEC 9            // (KCHUNK*FF)/(4*128) = (BSZ*KCHUNK)/(4*128) per-thread float4 transfers per chunk

__global__ __launch_bounds__(128) void lc2d_bn_wmma(
    const float* __restrict__ x, const float* __restrict__ gamma,
    const float* __restrict__ beta, const float* __restrict__ mean,
    const float* __restrict__ var, const float* __restrict__ weight,
    float* __restrict__ out)
{
    __shared__ __attribute__((aligned(16))) float lds_scale[CC];
    __shared__ __attribute__((aligned(16))) float lds_shift[CC];
    __shared__ __attribute__((aligned(16))) float ldsA[BSZ * AST];     // A[b][kk] BN-normalized patches
    __shared__ __attribute__((aligned(16))) float ldsW[KCHUNK * WST];  // W[kk][f] natural layout (async DMA target)

    const int pos  = blockIdx.x;          // 0..OHN*OWN-1
    const int h    = pos / OWN;
    const int w    = pos - h * OWN;
    const int tid  = threadIdx.x;         // 0..127
    const int lane = tid & 31;
    const int wave = tid >> 5;

    // Fold BN into per-channel scale/shift
    if (tid < CC) {
        float sc = gamma[tid] * rsqrtf(var[tid] + 1e-3f);
        lds_scale[tid] = sc;
        lds_shift[tid] = beta[tid] - mean[tid] * sc;
    }
    __syncthreads();

    const float* wg = weight + (size_t)pos * (KDIM * FF);

    // Wave tile assignment: 2x2 grid of 16x16 tiles over the 32x32 (batch x filter) output
    const int mt   = (wave >> 1) * 16;    // batch-tile base
    const int nt   = (wave & 1) * 16;     // filter-tile base
    const int row  = lane & 15;
    const int kh2  = (lane >> 4) * 2;     // lanes 16..31 take K+2,K+3

    const float* pa = &ldsA[(mt + row) * AST + kh2];       // A fragment base (b64 reads)
    const float* pb = &ldsW[kh2 * WST + (nt + row)];       // B fragment base (2x b32 reads, stride WST)

    v8f acc = {};

    for (int ck = 0; ck < KDIM; ck += KCHUNK) {
        // ---- async DMA: weight chunk, global -> LDS, no VGPR pass-through ----
        const float* wgc = wg + (size_t)ck * FF;
        #pragma unroll
        for (int u = 0; u < NVEC; ++u) {
            int e4 = 4 * (tid + u * 128);     // flat element offset in [k][f] chunk
            int k  = e4 >> 5;                 // /FF
            int f0 = e4 & 31;                 // 4-aligned -> 16B aligned on both sides
            __builtin_amdgcn_global_load_async_to_lds_b128(
                (v4i*)(wgc + e4),
                (v4i*)&ldsW[k * WST + f0], 0, 0);
        }

        // ---- patch chunk: batched vector loads, BN in registers, b128 LDS stores ----
        v4f xa[NVEC];
        int bb[NVEC], kks[NVEC];
        #pragma unroll
        for (int u = 0; u < NVEC; ++u) {
            int t2  = tid + u * 128;          // 0..1151
            int b   = t2 / 36;                // KCHUNK/4 = 36 float4 per batch row
            int kk  = 4 * (t2 - b * 36);
            int k   = ck + kk;                // K index = (i*KW + j)*CC + c
            int i   = k / (KW * CC);
            int rem = k - i * (KW * CC);
            int j   = rem >> 5;
            int c   = rem & 31;               // 4-aligned
            xa[u]  = *(const v4f*)&x[(((size_t)(b * HH + h + i)) * WW + (w + j)) * CC + c];
            bb[u]  = b;
            kks[u] = kk;
        }
        #pragma unroll
        for (int u = 0; u < NVEC; ++u) {
            int c = (ck + kks[u]) & 31;
            v4f sc = *(const v4f*)&lds_scale[c];
            v4f sh = *(const v4f*)&lds_shift[c];
            v4f r  = xa[u] * sc + sh;
            *(v4f*)&ldsA[bb[u] * AST + kks[u]] = r;
        }

        asm volatile("s_wait_asynccnt 0x0" ::: "memory");  // weight DMA landed in LDS
        __syncthreads();

        // ---- 36 fp32 WMMA steps over this K chunk ----
        #pragma unroll
        for (int k = 0; k < KCHUNK; k += 4) {
            v2f a = *(const v2f*)(pa + k);
            v2f b;
            b.x = pb[k * WST];
            b.y = pb[k * WST + WST];
            acc = __builtin_amdgcn_wmma_f32_16x16x4_f32(
                      false, a, false, b, (short)0, acc, false, false);
        }
        __syncthreads();   // before next chunk overwrites LDS
    }

    // ---- ReLU + store: C/D layout: VGPR r -> M = r + 8*(lane>=16), N = lane%16 ----
    const int f     = nt + row;
    const int mbase = mt + 8 * (lane >> 4);
    size_t idx = (((size_t)mbase * OHN + h) * OWN + w) * FF + f;
    const size_t bstride = (size_t)OHN * OWN * FF;   // stride between consecutive batches
    #pragma unroll
    for (int r = 0; r < 8; ++r) {
        float v = acc[r];
        out[idx] = v > 0.0f ? v : 0.0f;
        idx += bstride;
    }
}

extern "C" void kernel_launch(void* const* d_in, const int* in_sizes, int n_in,
                              void* d_out, int out_size, void* d_ws, size_t ws_size,
                              hipStream_t stream) {
    const float* x      = (const float*)d_in[0];
    const float* gamma  = (const float*)d_in[1];
    const float* beta   = (const float*)d_in[2];
    const float* mean   = (const float*)d_in[3];
    const float* var    = (const float*)d_in[4];
    const float* weight = (const float*)d_in[5];
    float* out = (float*)d_out;

    dim3 grid(OHN * OWN);
    dim3 block(128);
    hipLaunchKernelGGL(lc2d_bn_wmma, grid, block, 0, stream,
                       x, gamma, beta, mean, var, weight, out);
}